// GMABSE3_48902497632450
// MI455X (gfx1250) — compile-verified
//
#include <hip/hip_runtime.h>
#include <math.h>

typedef __attribute__((ext_vector_type(2))) float v2f;
typedef __attribute__((ext_vector_type(8))) float v8f;

// ---------------------------------------------------------------------------
// Phase 0: zero-init outputs (out0|out1 concatenated = N*64 floats) and the
// softmax denominator accumulator s (N*8 floats).
// ---------------------------------------------------------------------------
__global__ __launch_bounds__(256) void gma_zero(float* __restrict__ out, int n_out,
                                                float* __restrict__ s, int n_s) {
    int i = blockIdx.x * blockDim.x + threadIdx.x;
    if (i < n_out) out[i] = 0.0f;
    if (i < n_s)   s[i]   = 0.0f;
}

// ---------------------------------------------------------------------------
// Phase 1: per-edge, per-head attention scores via WMMA.
//
// One wave handles 2 edges = 16 edge-heads. Each edge-head has an 8-dim fiber
// (2 feats from *0 tensors, 6 feats from *1 tensors).
//   A (16x4 f32)  : rows = edge-heads, cols = feats f..f+3 of K
//   B (4x16 f32)  : rows = feats,      cols = edge-heads of Q[col[e]]
//   D = A*B + C   : diagonal D[i,i] = partial dot of edge-head i
// Two chained WMMAs cover feats 0..3 and 4..7.
// A layout (ISA 7.12.2): lanes 0-15 -> K=0(+f VGPR0),K=1(VGPR1); lanes 16-31 -> K=2,3.
// B layout mirrors C/D striping: VGPR0 holds rows {0,2}, VGPR1 rows {1,3}.
// Diagonal extraction: D[i,i] lives at VGPR (i&7), lane (i<8 ? i : i+16) ->
// bounce through LDS (same-wave DS ops are in-order; no barrier needed).
// ---------------------------------------------------------------------------
__global__ __launch_bounds__(256) void gma_phase1(const float* __restrict__ k0,
                                                  const float* __restrict__ k1,
                                                  const float* __restrict__ q0,
                                                  const float* __restrict__ q1,
                                                  const int*   __restrict__ edge_index,
                                                  float* __restrict__ e_ws,
                                                  float* __restrict__ s,
                                                  int E) {
    __shared__ float diagbuf[8][8][32];   // [wave][c-vgpr][lane]

    const int lane = threadIdx.x & 31;
    const int wave = threadIdx.x >> 5;
    const int pair = blockIdx.x * 8 + wave;
    const int e0   = pair * 2;
    if (e0 + 1 >= E) return;              // wave-uniform: EXEC stays all-ones

    const int M     = lane & 15;          // edge-head index 0..15
    const int fbase = (lane < 16) ? 0 : 2;
    const int edge  = e0 + (M >> 3);
    const int head  = M & 7;
    const int col   = edge_index[E + edge];   // edge_index[1][edge]

    const float* k0p = k0 + edge * 16 + head * 2;   // (E,16,1): feats 0,1
    const float* k1p = k1 + edge * 48 + head * 6;   // (E,16,3): feats 2..7
    const float* q0p = q0 + col  * 16 + head * 2;
    const float* q1p = q1 + col  * 48 + head * 6;

    auto kv = [&](int f) { return (f < 2) ? k0p[f] : k1p[f - 2]; };
    auto qv = [&](int f) { return (f < 2) ? q0p[f] : q1p[f - 2]; };

    v2f a1, a2, b1, b2;
    a1.x = kv(fbase);     a1.y = kv(fbase + 1);
    a2.x = kv(fbase + 4); a2.y = kv(fbase + 5);
    b1.x = qv(fbase);     b1.y = qv(fbase + 1);
    b2.x = qv(fbase + 4); b2.y = qv(fbase + 5);

    v8f c = {};
    c = __builtin_amdgcn_wmma_f32_16x16x4_f32(false, a1, false, b1, (short)0, c, false, false);
    c = __builtin_amdgcn_wmma_f32_16x16x4_f32(false, a2, false, b2, (short)0, c, false, false);

#pragma unroll
    for (int j = 0; j < 8; ++j) diagbuf[wave][j][lane] = c[j];
    __asm__ volatile("" ::: "memory");    // keep LDS store->load order

    if (lane < 16) {
        const int srcLane = (lane < 8) ? lane : (lane + 16);
        const float dot   = diagbuf[wave][lane & 7][srcLane];
        const float ev    = __expf(dot * 0.125f);   // 1/sqrt(64)
        e_ws[edge * 8 + head] = ev;
        const int r = edge_index[edge];             // edge_index[0][edge]
        atomicAdd(&s[r * 8 + head], ev);
    }
}

// ---------------------------------------------------------------------------
// Phase 2: a = e / s[row]; scatter a*v0 -> out0[col], a*v1 -> out1[col].
// One thread per (edge, m), m = 0..15; head = m>>1.  4 f32 atomics/thread,
// resolved at L2 (GLOBAL_ATOMIC_ADD_F32 without return).
// ---------------------------------------------------------------------------
__global__ __launch_bounds__(256) void gma_phase2(const float* __restrict__ v0,
                                                  const float* __restrict__ v1,
                                                  const int*   __restrict__ edge_index,
                                                  const float* __restrict__ e_ws,
                                                  const float* __restrict__ s,
                                                  float* __restrict__ out0,
                                                  float* __restrict__ out1,
                                                  int E) {
    const int tid = blockIdx.x * blockDim.x + threadIdx.x;
    if (tid >= E * 16) return;
    const int e = tid >> 4;
    const int m = tid & 15;
    const int h = m >> 1;

    const int r  = edge_index[e];        // row: softmax denominator segment
    const int cn = edge_index[E + e];    // col: output segment

    const float a = e_ws[e * 8 + h] / s[r * 8 + h];

    atomicAdd(&out0[cn * 16 + m], a * v0[e * 16 + m]);

    const float* v1p = v1 + (e * 16 + m) * 3;
    float*       o1p = out1 + (cn * 16 + m) * 3;
    atomicAdd(o1p + 0, a * v1p[0]);
    atomicAdd(o1p + 1, a * v1p[1]);
    atomicAdd(o1p + 2, a * v1p[2]);
}

// ---------------------------------------------------------------------------
// Inputs (setup_inputs order): v0, v1, k0, k1, q0, q1, edge_index
// d_out: out0 (N*16) | out1 (N*48) floats.
// d_ws : s (N*8 f32) | e_ws (E*8 f32)  => ~27.2 MB.
// ---------------------------------------------------------------------------
extern "C" void kernel_launch(void* const* d_in, const int* in_sizes, int n_in,
                              void* d_out, int out_size, void* d_ws, size_t ws_size,
                              hipStream_t stream) {
    const float* v0 = (const float*)d_in[0];
    const float* v1 = (const float*)d_in[1];
    const float* k0 = (const float*)d_in[2];
    const float* k1 = (const float*)d_in[3];
    const float* q0 = (const float*)d_in[4];
    const float* q1 = (const float*)d_in[5];
    const int* edge_index = (const int*)d_in[6];

    const int E = in_sizes[6] / 2;        // 800000
    const int N = in_sizes[4] / 16;       // 50000

    float* out0 = (float*)d_out;
    float* out1 = out0 + (size_t)N * 16;

    float* s    = (float*)d_ws;           // N*8 floats
    float* e_ws = s + (size_t)N * 8;      // E*8 floats

    // Phase 0: zero outputs + denominators
    const int n_out = N * 64;
    gma_zero<<<(n_out + 255) / 256, 256, 0, stream>>>(out0, n_out, s, N * 8);

    // Phase 1: WMMA scores + exp + segment-sum(e, row)
    const int pairs = (E + 1) / 2;
    gma_phase1<<<(pairs + 7) / 8, 256, 0, stream>>>(k0, k1, q0, q1, edge_index,
                                                    e_ws, s, E);

    // Phase 2: normalize + scatter into out[col]
    const int total = E * 16;
    gma_phase2<<<(total + 255) / 256, 256, 0, stream>>>(v0, v1, edge_index,
                                                        e_ws, s, out0, out1, E);
}